// seq_rewriter_85529978733206
// MI455X (gfx1250) — compile-verified
//
#include <hip/hip_runtime.h>
#include <cstdint>

// Problem constants (B=64, S=512, T=4096, V=4096, FW=5)
#define B_DIM 64
#define S_DIM 512
#define T_DIM 4096
#define V_DIM 4096
#define FW    5
#define TC    128                 // t-chunk width staged in LDS per step
#define NCHUNK (T_DIM / TC)       // 32
#define NROWS 80                  // 5 k-taps * 16 positions -> gathered rows per tile
#define RS    132                 // padded LDS row stride in floats (kills 2-way bank conflict)

typedef __attribute__((ext_vector_type(2))) float v2f;   // f32 WMMA A/B frag (16x4 / 4x16)
typedef __attribute__((ext_vector_type(8))) float v8f;   // f32 WMMA C/D frag (16x16)
typedef __attribute__((ext_vector_type(4))) int   v4i;

typedef __attribute__((address_space(1))) v4i GV4I;      // global int4
typedef __attribute__((address_space(3))) v4i LV4I;      // LDS int4

// ---- CDNA5 async global->LDS copy (16B/lane), ASYNCcnt-tracked -------------
__device__ __forceinline__ void async_copy_b128(const float* g, float* l) {
#if __has_builtin(__builtin_amdgcn_global_load_async_to_lds_b128)
  __builtin_amdgcn_global_load_async_to_lds_b128(
      (GV4I*)(uintptr_t)g, (LV4I*)(uint32_t)(uintptr_t)l, 0, 0);
#else
  asm volatile("global_load_async_to_lds_b128 %0, %1, off"
               :: "v"((uint32_t)(uintptr_t)l), "v"(g) : "memory");
#endif
}

__device__ __forceinline__ void wait_asynccnt_10() {
#if __has_builtin(__builtin_amdgcn_s_wait_asynccnt)
  __builtin_amdgcn_s_wait_asynccnt(10);
#else
  asm volatile("s_wait_asynccnt 0xA" ::: "memory");
#endif
}
__device__ __forceinline__ void wait_asynccnt_0() {
#if __has_builtin(__builtin_amdgcn_s_wait_asynccnt)
  __builtin_amdgcn_s_wait_asynccnt(0);
#else
  asm volatile("s_wait_asynccnt 0x0" ::: "memory");
#endif
}

// ---- Pass 1: W(T,V,FW) f32 -> W2[k][v][t] so rows are contiguous in t ------
__global__ __launch_bounds__(256) void transpose_W_kernel(
    const float* __restrict__ W, float* __restrict__ W2) {
  __shared__ float tile[FW][32][33];
  const int t0 = blockIdx.x * 32, v0 = blockIdx.y * 32;
  const int tid = threadIdx.x;
  const int vv = tid & 31, tq = tid >> 5;          // 8 t-rows per pass
#pragma unroll
  for (int it = 0; it < 4; ++it) {
    int tt = tq + it * 8;
    const float* src = W + (size_t)(t0 + tt) * (V_DIM * FW) + (size_t)(v0 + vv) * FW;
#pragma unroll
    for (int k = 0; k < FW; ++k) tile[k][vv][tt] = src[k];
  }
  __syncthreads();
  for (int e = tid; e < FW * 32 * 32; e += 256) {   // coalesced writes along t
    int k = e >> 10, rem = e & 1023, v2i = rem >> 5, t2i = rem & 31;
    W2[((size_t)(k * V_DIM + v0 + v2i) << 12) + (t0 + t2i)] = tile[k][v2i][t2i];
  }
}

// ---- Pass 2: 1 block = 16 positions; one-hot-GEMM logits via f32 WMMA ------
// Row j (j = k*16 + pos) of the staged B matrix = W2 row for token at
// p = st + pos + k - 2.  A[pos][j] = valid(p) ? 1 : 0 (banded one-hot mask).
// logits(16 x 16cols) accumulated over 20 chained V_WMMA_F32_16X16X4_F32.
__global__ __launch_bounds__(256) void seq_logits_kernel(
    const float* __restrict__ W2, const float* __restrict__ bvec,
    const int* __restrict__ toks, float* __restrict__ out,
    float* __restrict__ wsH) {
  extern __shared__ char smem[];
  float* lds0 = (float*)smem;                                 // 2 * NROWS * RS floats
  float4* red = (float4*)(smem + (size_t)2 * NROWS * RS * 4); // 8 waves * 16 pos

  const int tid  = threadIdx.x;
  const int w    = tid >> 5, lane = tid & 31;
  const int M    = lane & 15;                 // A-row / B-col / D-col index
  const int K0   = (lane < 16) ? 0 : 2;       // K slot in VGPR0 per ISA layout
  const int bidx = blockIdx.x >> 5;
  const int st   = (blockIdx.x & 31) << 4;    // tile start position
  const int t0   = w << 4;                    // this wave's 16-col slice in chunk

  // This wave's 10 gathered-row base pointers (reused across all 32 chunks).
  const float* rowptr[10];
#pragma unroll
  for (int i = 0; i < 10; ++i) {
    int j = w * 10 + i;
    int k = j >> 4, pos = j & 15;
    int p = st + pos + k - 2;
    int tok = (p >= 0 && p < S_DIM) ? toks[bidx * S_DIM + p] : 0;
    rowptr[i] = W2 + ((size_t)(k * V_DIM + tok) << 12);
  }

  // A fragments: 20 K=4 slices of the 16x80 banded selection matrix.
  float amask[FW];
#pragma unroll
  for (int k = 0; k < FW; ++k) {
    int p = st + M + k - 2;
    amask[k] = (p >= 0 && p < S_DIM) ? 1.0f : 0.0f;
  }
  v2f afr[20];
#pragma unroll
  for (int c = 0; c < 20; ++c) {
    int j0 = 4 * c + K0;
    int j1 = j0 + 1;
    int kc = c >> 2;                          // == j0>>4 == j1>>4 (K0 < 4)
    afr[c].x = ((j0 & 15) == M) ? amask[kc] : 0.0f;
    afr[c].y = ((j1 & 15) == M) ? amask[kc] : 0.0f;
  }

  // Online softmax state: slot r <-> position (r + 8*(lane>>4)), col = M.
  float sm[8], sl[8], ss[8];
  int   si[8];
#pragma unroll
  for (int r = 0; r < 8; ++r) {
    sm[r] = -__builtin_inff(); sl[r] = 0.f; ss[r] = 0.f; si[r] = 0;
  }

  // Prologue: stage chunk 0 (each wave: 10 async b128 ops = its 10 rows).
  {
    int coff = lane * 4;
#pragma unroll
    for (int i = 0; i < 10; ++i) {
      int j = w * 10 + i;
      async_copy_b128(rowptr[i] + coff, lds0 + j * RS + lane * 4);
    }
  }

  for (int n = 0; n < NCHUNK; ++n) {
    if (n + 1 < NCHUNK) {                     // prefetch next chunk (double buffer)
      float* buf = lds0 + ((n + 1) & 1) * (NROWS * RS);
      int coff = (n + 1) * TC + lane * 4;
#pragma unroll
      for (int i = 0; i < 10; ++i) {
        int j = w * 10 + i;
        async_copy_b128(rowptr[i] + coff, buf + j * RS + lane * 4);
      }
      wait_asynccnt_10();                     // in-order: chunk n fully landed
    } else {
      wait_asynccnt_0();
    }
    __syncthreads();                          // everyone's chunk-n rows visible

    const float* buf = lds0 + (n & 1) * (NROWS * RS);
    const int tcol = n * TC + t0 + M;
    const float bb = bvec[tcol];
    v8f d = {bb, bb, bb, bb, bb, bb, bb, bb}; // C init = bias broadcast per column
#pragma unroll
    for (int c = 0; c < 20; ++c) {
      v2f bf;                                 // B 4x16: VGPR0 = K{0,2}, VGPR1 = K{1,3}
      bf.x = buf[(4 * c + K0) * RS + t0 + M];
      bf.y = buf[(4 * c + K0 + 1) * RS + t0 + M];
      d = __builtin_amdgcn_wmma_f32_16x16x4_f32(false, afr[c], false, bf,
                                                (short)0, d, false, false);
    }
#pragma unroll
    for (int r = 0; r < 8; ++r) {             // flash-style online update
      float v = d[r];
      if (v > sm[r]) {
        float e = __expf(sm[r] - v);
        sl[r] = sl[r] * e + 1.0f;
        ss[r] = ss[r] * e + v;
        sm[r] = v; si[r] = tcol;
      } else {
        float e = __expf(v - sm[r]);
        sl[r] += e;
        ss[r] += e * v;
      }
    }
    __syncthreads();                          // chunk-n buffer free for reuse
  }

  // Reduce the 16 columns of each half-wave (xor stays inside each half).
#pragma unroll
  for (int r = 0; r < 8; ++r) {
    float m = sm[r], l = sl[r], s2 = ss[r]; int idx = si[r];
#pragma unroll
    for (int off = 8; off >= 1; off >>= 1) {
      float om = __shfl_xor(m, off);
      float ol = __shfl_xor(l, off);
      float os = __shfl_xor(s2, off);
      int   oi = __shfl_xor(idx, off);
      bool take = (om > m) || (om == m && oi < idx);
      if (take) { float e = __expf(m - om); l = l * e + ol; s2 = s2 * e + os; m = om; idx = oi; }
      else      { float e = __expf(om - m); l += ol * e; s2 += os * e; }
    }
    if (M == 0) {
      int pos = r + ((lane >> 4) << 3);
      red[w * 16 + pos] = make_float4(m, l, s2, (float)idx);
    }
  }
  __syncthreads();

  // Merge the 8 waves in fixed order (deterministic), emit outputs.
  if (tid < 16) {
    int pos = tid;
    float m = -__builtin_inff(), l = 0.f, s2 = 0.f; int idx = 0x7FFFFFFF;
    for (int ww = 0; ww < 8; ++ww) {
      float4 v = red[ww * 16 + pos];
      float om = v.x, ol = v.y, os = v.z; int oi = (int)v.w;
      bool take = (om > m) || (om == m && oi < idx);
      if (take) { float e = __expf(m - om); l = l * e + ol; s2 = s2 * e + os; m = om; idx = oi; }
      else      { float e = __expf(om - m); l += ol * e; s2 += os * e; }
    }
    int o = bidx * S_DIM + st + pos;
    float lg = __logf(l);
    out[o] = (float)idx;                      // new_seq (argmax over T)
    out[B_DIM * S_DIM + o] = -lg;             // log_p at argmax = -(log sum exp(x-m))
    float H = (m + lg) - s2 / l;              // lse - sum(p*logit)
#pragma unroll
    for (int off = 8; off >= 1; off >>= 1) H += __shfl_xor(H, off);
    if (pos == 0) wsH[blockIdx.x] = H;
  }
}

// ---- Pass 3: deterministic fixed-order entropy reduction -------------------
__global__ __launch_bounds__(256) void reduce_entropy_kernel(
    const float* __restrict__ wsH, float* __restrict__ out) {
  __shared__ float sb[256];
  float acc = 0.f;
  for (int i = threadIdx.x; i < (B_DIM * S_DIM / 16); i += 256) acc += wsH[i];
  sb[threadIdx.x] = acc;
  __syncthreads();
  for (int o = 128; o > 0; o >>= 1) {
    if ((int)threadIdx.x < o) sb[threadIdx.x] += sb[threadIdx.x + o];
    __syncthreads();
  }
  if (threadIdx.x == 0)
    out[0] = sb[0] * (1.0f / ((float)B_DIM * (float)S_DIM * (float)T_DIM));
}

extern "C" void kernel_launch(void* const* d_in, const int* in_sizes, int n_in,
                              void* d_out, int out_size, void* d_ws, size_t ws_size,
                              hipStream_t stream) {
  const float* W = (const float*)d_in[0];          // (T, V, FW) f32
  const float* b = (const float*)d_in[1];          // (T,) f32
  const int*   x = (const int*)d_in[2];            // (B, S) int32
  float* out = (float*)d_out;                      // [new_seq | log_probs | entropy]

  float* W2  = (float*)d_ws;                       // FW*V*T f32 = 335.5 MB
  float* wsH = (float*)((char*)d_ws + (size_t)FW * V_DIM * T_DIM * sizeof(float));

  // Pass 1: layout transform (runs every call; no cross-call state allowed).
  transpose_W_kernel<<<dim3(T_DIM / 32, V_DIM / 32), 256, 0, stream>>>(W, W2);

  // Pass 2: 2048 tiles of 16 positions, 8 waves each, 86.5 KB dynamic LDS.
  size_t shbytes = (size_t)2 * NROWS * RS * sizeof(float) + 8 * 16 * sizeof(float4);
  (void)hipFuncSetAttribute((const void*)seq_logits_kernel,
                            hipFuncAttributeMaxDynamicSharedMemorySize, (int)shbytes);
  seq_logits_kernel<<<(B_DIM * S_DIM) / 16, 256, shbytes, stream>>>(W2, b, x, out, wsH);

  // Pass 3: entropy scalar.
  reduce_entropy_kernel<<<1, 256, 0, stream>>>(wsH, out + 2 * B_DIM * S_DIM);
}